// ScaledDotProductAttention_44341242364479
// MI455X (gfx1250) — compile-verified
//
#include <hip/hip_runtime.h>
#include <hip/hip_bf16.h>

// Attention with hard-argmax softmax (temperature=1e-14) on gfx1250.
// B=16, N=2048, D=64. Outputs: attn [B,N,N] then out [B,N,D], concatenated.

typedef __attribute__((ext_vector_type(2))) float v2f;
typedef __attribute__((ext_vector_type(8))) float v8f;

#define BATCH 16
#define SEQ   2048
#define HDIM  64
#define SCALE_INV 0.125f   // 1/8
#define NTHREADS 128       // 4 wave32s
#define COLTILES (SEQ / 16)  // 128

__global__ __launch_bounds__(NTHREADS, 1)
void attn_hardmax_kernel(const float* __restrict__ q,
                         const float* __restrict__ kmat,
                         const float* __restrict__ v,
                         const unsigned char* __restrict__ mask,
                         float* __restrict__ attn_out,
                         float* __restrict__ o_out) {
    // Score strip for 16 query rows x full key length, fp32, kept in LDS.
    // 16 * 2052 * 4 B = 131,328 B  (CDNA5 WGP has 320 KB LDS)
    __shared__ float S[16][SEQ + 4];
    __shared__ float pmax[16][8];
    __shared__ int   pcnt[16][8];
    __shared__ float rmax[16];
    __shared__ float rinv[16];

    const int tid  = threadIdx.x;
    const int lane = tid & 31;
    const int wave = tid >> 5;          // 0..3
    const int r    = lane & 15;         // row-in-tile / col-in-tile index
    const int hb   = lane >> 4;         // half-wave selector

    const int b  = blockIdx.y;
    const int m0 = blockIdx.x * 16;     // query row tile base

    // ---- Phase 1: S = mask ? -inf : (Q K^T)/8 via V_WMMA_F32_16X16X4_F32 ----
    // A (16x4 f32) layout: lanes 0-15 hold {K=0,K=1}, lanes 16-31 hold {K=2,K=3}
    // per 4-wide K step => per lane a float2 at offset 4*step + hb*2.
    v2f aq[16];
    {
        const float* qp = q + ((size_t)(b * SEQ + m0 + r)) * HDIM + hb * 2;
        #pragma unroll
        for (int s = 0; s < 16; ++s)
            aq[s] = *(const v2f*)(qp + s * 4);
    }

    for (int jt = wave; jt < COLTILES; jt += 4) {
        const int j0 = jt * 16;
        // B = K^T (4x16): lane's column n = r, so lane loads K[j0+r][...],
        // same element pattern as A.
        v2f bk[16];
        const float* kp = kmat + ((size_t)(b * SEQ + j0 + r)) * HDIM + hb * 2;
        #pragma unroll
        for (int s = 0; s < 16; ++s)
            bk[s] = *(const v2f*)(kp + s * 4);

        v8f c = {};
        #pragma unroll
        for (int s = 0; s < 16; ++s)
            c = __builtin_amdgcn_wmma_f32_16x16x4_f32(
                    false, aq[s], false, bk[s], (short)0, c, false, false);

        // C layout: VGPR g -> row m = g + 8*hb, col n = r.
        const size_t mrow_base = ((size_t)b * SEQ + m0) * (size_t)SEQ;
        #pragma unroll
        for (int g = 0; g < 8; ++g) {
            const int mr = g + hb * 8;
            const unsigned char mv =
                mask[mrow_base + (size_t)mr * SEQ + (size_t)(j0 + r)];
            const float sval = mv ? -__builtin_inff() : c[g] * SCALE_INV;
            S[mr][j0 + r] = sval;
        }
    }
    __syncthreads();

    // ---- Phase 2: row max + tie count (online) ----
    {
        const int row = tid & 15;
        const int seg = tid >> 4;       // 8 segments of 256 columns
        const int c0  = seg * 256;
        float mx = -__builtin_inff();
        int   cn = 0;
        for (int n = 0; n < 256; ++n) {
            const float s = S[row][c0 + n];
            if (s > mx) { mx = s; cn = 1; }
            else if (s == mx) { ++cn; }
        }
        pmax[row][seg] = mx;
        pcnt[row][seg] = cn;
    }
    __syncthreads();
    if (tid < 16) {
        float g = -__builtin_inff();
        #pragma unroll
        for (int s = 0; s < 8; ++s) g = fmaxf(g, pmax[tid][s]);
        int tot = 0;
        #pragma unroll
        for (int s = 0; s < 8; ++s)
            if (pmax[tid][s] == g) tot += pcnt[tid][s];
        if (tot < 1) tot = 1;
        rmax[tid] = g;
        rinv[tid] = 1.0f / (float)tot;
    }
    __syncthreads();

    // ---- Phase 3: stream one-hot attn rows to global (coalesced) ----
    {
        const size_t base = ((size_t)b * SEQ + m0) * (size_t)SEQ;
        for (int idx = tid; idx < 16 * SEQ; idx += NTHREADS) {
            const int row = idx >> 11;      // / 2048
            const int col = idx & (SEQ - 1);
            const float s = S[row][col];
            attn_out[base + (size_t)row * SEQ + col] =
                (s == rmax[row]) ? rinv[row] : 0.0f;
        }
    }

    // ---- Phase 4: out = attn @ V as a gather-average of tied V rows ----
    {
        const int orow = tid >> 3;          // 16 rows
        const int d0   = (tid & 7) * 8;     // 8 threads cover D=64
        const float gmax = rmax[orow];
        float acc[8];
        #pragma unroll
        for (int d = 0; d < 8; ++d) acc[d] = 0.0f;
        for (int n = 0; n < SEQ; ++n) {
            const float s = S[orow][n];     // LDS broadcast across 8 threads
            if (s == gmax) {
                const float* vp = v + ((size_t)b * SEQ + n) * HDIM + d0;
                #pragma unroll
                for (int d = 0; d < 8; ++d) acc[d] += vp[d];
            }
        }
        float* op = o_out + ((size_t)b * SEQ + m0 + orow) * HDIM + d0;
        const float inv = rinv[orow];
        #pragma unroll
        for (int d = 0; d < 8; ++d) op[d] = acc[d] * inv;
    }
}

extern "C" void kernel_launch(void* const* d_in, const int* in_sizes, int n_in,
                              void* d_out, int out_size, void* d_ws, size_t ws_size,
                              hipStream_t stream) {
    const float* q = (const float*)d_in[0];
    const float* k = (const float*)d_in[1];
    const float* v = (const float*)d_in[2];
    const unsigned char* mask = (const unsigned char*)d_in[3]; // jnp.bool_ -> 1 byte

    float* attn = (float*)d_out;                                   // [B,N,N]
    float* outp = attn + (size_t)BATCH * SEQ * SEQ;                // [B,N,D]

    dim3 grid(SEQ / 16, BATCH);   // 128 x 16 workgroups
    dim3 block(NTHREADS);
    attn_hardmax_kernel<<<grid, block, 0, stream>>>(q, k, v, mask, attn, outp);
}